// GraphSAGELayerPrimitive_41807211659455
// MI455X (gfx1250) — compile-verified
//
#include <hip/hip_runtime.h>
#include <hip/hip_bf16.h>

// CDNA5 / gfx1250 GraphSAGE mean-agg layer.
// out = relu( (scatter_mean(x[src] -> dst)) @ W_l + x @ W_r + b_l + b_r )

typedef __attribute__((ext_vector_type(2))) float v2f;
typedef __attribute__((ext_vector_type(8))) float v8f;

#define F_DIM 128   // F_IN == F_OUT == 128

// ---------------------------------------------------------------------------
// Kernel 1: zero the workspace (sum_agg [N*128] + cnt [N]) with float4 stores.
// ---------------------------------------------------------------------------
__global__ __launch_bounds__(256) void zero_ws_kernel(float4* __restrict__ p,
                                                      long long n4) {
    long long i = (long long)blockIdx.x * blockDim.x + threadIdx.x;
    long long stride = (long long)gridDim.x * blockDim.x;
    float4 z = make_float4(0.f, 0.f, 0.f, 0.f);
    for (; i < n4; i += stride) p[i] = z;
}

// ---------------------------------------------------------------------------
// Kernel 2: per-edge gather + scatter-add. One wave32 per edge.
// Lane l moves x[src][4l..4l+3] (coalesced 512B per wave) and atomically adds
// into sum_agg[dst][4l..4l+3]. Lane 0 bumps the in-degree counter.
// Relaxed agent-scope float adds -> global_atomic_add_f32 (L2-resident RMW).
// ---------------------------------------------------------------------------
__global__ __launch_bounds__(256) void sage_scatter_kernel(
    const long long* __restrict__ edge_index,   // [2, E] int64
    const float*     __restrict__ x,            // [N, 128]
    float*           __restrict__ sum_agg,      // [N, 128]
    float*           __restrict__ cnt,          // [N]
    int n_edges) {
    const int wave = threadIdx.x >> 5;
    const int lane = threadIdx.x & 31;
    const long long e = (long long)blockIdx.x * 8 + wave;
    if (e >= n_edges) return;   // wave-uniform branch

    const int src = (int)edge_index[e];
    const int dst = (int)edge_index[(long long)n_edges + e];

    const float4 v = *(const float4*)(x + (size_t)src * F_DIM + lane * 4);
    float* d = sum_agg + (size_t)dst * F_DIM + lane * 4;
    __hip_atomic_fetch_add(d + 0, v.x, __ATOMIC_RELAXED, __HIP_MEMORY_SCOPE_AGENT);
    __hip_atomic_fetch_add(d + 1, v.y, __ATOMIC_RELAXED, __HIP_MEMORY_SCOPE_AGENT);
    __hip_atomic_fetch_add(d + 2, v.z, __ATOMIC_RELAXED, __HIP_MEMORY_SCOPE_AGENT);
    __hip_atomic_fetch_add(d + 3, v.w, __ATOMIC_RELAXED, __HIP_MEMORY_SCOPE_AGENT);
    if (lane == 0)
        __hip_atomic_fetch_add(cnt + dst, 1.0f, __ATOMIC_RELAXED, __HIP_MEMORY_SCOPE_AGENT);
}

// ---------------------------------------------------------------------------
// Kernel 3: fused mean + dual GEMM + bias + relu via V_WMMA_F32_16X16X4_F32.
// Block = 256 threads = 8 wave32s -> 128 rows x 128 cols of output.
// Both weight matrices are staged in ONE interleaved 128 KB LDS array
// (WGP has 320 KB), PRE-SWIZZLED into per-WMMA-fragment order so each lane's
// B fragment is a single conflict-free ds_load_b64 into an even VGPR pair,
// and all 16 fragments of a k-step live within a 4 KB window so every DS
// access is an immediate offset off one rolling base register:
//   slot[(kb*16 + t*2 + which)*32 + lane] = { W[k][n], W[k+1][n] }
//   which: 0 = W_l, 1 = W_r;  k = kb*4 + (lane>=16 ? 2 : 0);  n = t*16+(lane&15)
// Single fused k-loop: 2 global_load_b64 (A frags) + 16 ds_load_b64 + 16 WMMA.
// WMMA f32 16x16x4 fragment layout (ISA 7.12.2):
//   A (16x4):  lane<16 -> M=lane, K={0,1}; lane>=16 -> M=lane-16, K={2,3}
//   B (4x16):  lane<16 -> N=lane, K={0,1}; lane>=16 -> N=lane-16, K={2,3}
//   C (16x16): vgpr r,  lane<16 -> M=r, N=lane; lane>=16 -> M=r+8, N=lane-16
// ---------------------------------------------------------------------------
__global__ __launch_bounds__(256) void sage_gemm_kernel(
    const float* __restrict__ sum_agg,  // [N, 128] (scatter result)
    const float* __restrict__ cnt,      // [N]
    const float* __restrict__ x,        // [N, 128]
    const float* __restrict__ Wl,       // [128, 128] row-major [in, out]
    const float* __restrict__ bl,       // [128]
    const float* __restrict__ Wr,       // [128, 128]
    const float* __restrict__ br,       // [128]
    float*       __restrict__ out,      // [N, 128]
    int n_nodes) {
    // 16384 fragments (32 k-steps x 8 col-tiles x 2 matrices x 32 lanes), 8B each
    __shared__ v2f sW[16384];   // 128 KB

    const int tid  = threadIdx.x;
    const int wave = tid >> 5;
    const int lane = tid & 31;
    const int half = lane >> 4;       // 0: lanes 0-15, 1: lanes 16-31
    const int m    = lane & 15;       // row-in-tile (A) / col-in-tile (B,C)
    const int koff = half * 2;        // this lane's K sub-offset {0,2}
    const int row0 = blockIdx.x * 128 + wave * 16;

    // --- swizzled cooperative fill (one-time; W is L2-resident across blocks)
    for (int s = tid; s < 16384; s += 256) {
        const int kb    = s >> 9;          // k-step        (512 slots per kb)
        const int sub   = (s >> 5) & 15;   // t*2 + which
        const int t     = sub >> 1;
        const int which = sub & 1;
        const int l     = s & 31;          // target lane
        const int k     = kb * 4 + ((l & 16) ? 2 : 0);
        const int n     = t * 16 + (l & 15);
        const float* W  = which ? Wr : Wl;
        v2f f;
        f.x = W[k * F_DIM + n];
        f.y = W[(k + 1) * F_DIM + n];
        sW[s] = f;
    }
    __syncthreads();

    // per-lane A-row (clamped for the grid tail; stores are predicated later)
    int rowA = row0 + m;
    if (rowA >= n_nodes) rowA = n_nodes - 1;
    const float* sumRow = sum_agg + (size_t)rowA * F_DIM;
    const float* xRow   = x       + (size_t)rowA * F_DIM;
    const float  inv    = 1.0f / fmaxf(cnt[rowA], 1.0f);   // fused mean

    v8f acc[8] = {};   // 8 col-tiles x (16x16 f32) = full 16x128 row strip

    // --- fused dual-GEMM k-loop: C += mean*Wl + x*Wr ---
    for (int kb = 0; kb < 32; ++kb) {
        const int off = kb * 4 + koff;
        v2f am = *(const v2f*)(sumRow + off);
        am.x *= inv;
        am.y *= inv;
        const v2f ax = *(const v2f*)(xRow + off);
        const int base = (kb << 9) + lane;   // rolling 4 KB window per k-step
        #pragma unroll
        for (int t = 0; t < 8; ++t) {
            const v2f bfl = sW[base + (t << 6)];
            acc[t] = __builtin_amdgcn_wmma_f32_16x16x4_f32(
                false, am, false, bfl, (short)0, acc[t], false, false);
            const v2f bfr = sW[base + (t << 6) + 32];
            acc[t] = __builtin_amdgcn_wmma_f32_16x16x4_f32(
                false, ax, false, bfr, (short)0, acc[t], false, false);
        }
    }

    // --- bias + relu + predicated store (C layout: M = r + half*8, N = n) ---
    #pragma unroll
    for (int t = 0; t < 8; ++t) {
        const int n = t * 16 + m;
        const float bias = bl[n] + br[n];
        #pragma unroll
        for (int r = 0; r < 8; ++r) {
            const int row = row0 + r + half * 8;
            if (row < n_nodes) {
                float v = acc[t][r] + bias;
                out[(size_t)row * F_DIM + n] = v > 0.f ? v : 0.f;
            }
        }
    }
}

// ---------------------------------------------------------------------------
// Launch: zero ws -> edge scatter -> fused mean/GEMM/relu
// Inputs (setup_inputs order): x, edge_index(int64), W_l, b_l, W_r, b_r
// ---------------------------------------------------------------------------
extern "C" void kernel_launch(void* const* d_in, const int* in_sizes, int n_in,
                              void* d_out, int out_size, void* d_ws, size_t ws_size,
                              hipStream_t stream) {
    const float*     x  = (const float*)d_in[0];
    const long long* ei = (const long long*)d_in[1];
    const float*     Wl = (const float*)d_in[2];
    const float*     bl = (const float*)d_in[3];
    const float*     Wr = (const float*)d_in[4];
    const float*     br = (const float*)d_in[5];
    float* out = (float*)d_out;

    const int n_nodes = in_sizes[0] / F_DIM;
    const int n_edges = in_sizes[1] / 2;

    float* sum_agg = (float*)d_ws;                       // N*128 floats
    float* cnt     = sum_agg + (size_t)n_nodes * F_DIM;  // N floats

    // zero sum_agg + cnt (N*129 floats, divisible by 4)
    const long long n4 = ((long long)n_nodes * (F_DIM + 1)) / 4;
    zero_ws_kernel<<<2048, 256, 0, stream>>>((float4*)d_ws, n4);

    // one wave32 per edge, 8 edges per 256-thread block
    const int sblocks = (n_edges + 7) / 8;
    sage_scatter_kernel<<<sblocks, 256, 0, stream>>>(ei, x, sum_agg, cnt, n_edges);

    // 128 rows per block
    const int gblocks = (n_nodes + 127) / 128;
    sage_gemm_kernel<<<gblocks, 256, 0, stream>>>(sum_agg, cnt, x, Wl, bl, Wr, br,
                                                  out, n_nodes);
}